// MLPPredictor_893353198523
// MI455X (gfx1250) — compile-verified
//
#include <hip/hip_runtime.h>
#include <hip/hip_bf16.h>

// CDNA5 / gfx1250. wave32.
typedef __attribute__((ext_vector_type(2))) float v2f;
typedef __attribute__((ext_vector_type(8))) float v8f;

#define D_FEAT 128
#define HIDDEN 128

// ---------------------------------------------------------------------------
// Kernel 0: transpose/fuse W1 ([128 x 256], row k = output unit) into
// Wt[d][n'] (128 x 256) so that B-fragment loads are coalesced across lanes:
//   n' in [0,128):   Wt[d][n']   = W1[n'][d]        (W1a^T)
//   n' in [128,256): Wt[d][n']   = W1[n'-128][128+d] (W1b^T)
// ---------------------------------------------------------------------------
__global__ void mlp_prep_wt(const float* __restrict__ W1, float* __restrict__ Wt) {
    int idx = blockIdx.x * blockDim.x + threadIdx.x;
    if (idx >= D_FEAT * 256) return;
    int d = idx >> 8;   // 0..127
    int n = idx & 255;  // 0..255
    float v = (n < HIDDEN) ? W1[n * (2 * D_FEAT) + d]
                           : W1[(n - HIDDEN) * (2 * D_FEAT) + D_FEAT + d];
    Wt[idx] = v;
}

// ---------------------------------------------------------------------------
// Kernel 1: AB[n][0..255] = h[n] @ Wt   (fp32 WMMA 16x16x4)
// Block = 256 threads = 8 waves; one 16-node tile per block; wave w owns
// output column tiles {2w, 2w+1}. K loop: 128 / 4 = 32 steps, 2 WMMA each.
// ---------------------------------------------------------------------------
__global__ void mlp_node_gemm(const float* __restrict__ h,
                              const float* __restrict__ Wt,
                              float* __restrict__ AB,
                              int n_nodes) {
    const int wave  = threadIdx.x >> 5;   // 0..7
    const int lane  = threadIdx.x & 31;
    const int l16   = lane & 15;
    const int khalf = lane >> 4;          // 0 or 1
    const int m0    = blockIdx.x * 16;

    int row = m0 + l16;
    int rowc = row < n_nodes ? row : (n_nodes - 1);   // clamp reads, no divergence
    const float* __restrict__ hrow = h + (size_t)rowc * D_FEAT;

    const int n0   = wave * 32;           // first of two 16-wide column tiles
    const int ncol = n0 + l16;

    v8f acc0 = {};
    v8f acc1 = {};

    for (int k0 = 0; k0 < D_FEAT; k0 += 4) {
        int k = k0 + 2 * khalf;
        // A fragment (16x4 fp32): lanes 0-15 -> K = k0,k0+1 ; lanes 16-31 -> K = k0+2,k0+3
        v2f a;
        a.x = hrow[k];
        a.y = hrow[k + 1];
        // B fragments (4x16 fp32), mirrored layout, coalesced over lanes
        v2f b0, b1;
        b0.x = Wt[(size_t)k * 256 + ncol];
        b0.y = Wt[(size_t)(k + 1) * 256 + ncol];
        b1.x = Wt[(size_t)k * 256 + ncol + 16];
        b1.y = Wt[(size_t)(k + 1) * 256 + ncol + 16];

        acc0 = __builtin_amdgcn_wmma_f32_16x16x4_f32(false, a, false, b0,
                                                     (short)0, acc0, false, false);
        acc1 = __builtin_amdgcn_wmma_f32_16x16x4_f32(false, a, false, b1,
                                                     (short)0, acc1, false, false);
    }

    // D layout: VGPR r, lane l -> M = r + 8*(l>=16), N = l%16
    const int mbase = m0 + 8 * khalf;
    for (int r = 0; r < 8; ++r) {
        int m = mbase + r;
        if (m < n_nodes) {
            AB[(size_t)m * 256 + ncol]      = acc0[r];
            AB[(size_t)m * 256 + ncol + 16] = acc1[r];
        }
    }
}

// ---------------------------------------------------------------------------
// Kernel 2: per-edge epilogue. One wave per edge; lane i handles 4 hidden
// units. score = sum_k relu(A[src][k] + B[dst][k] + b1[k]) * W2[k] + b2.
// A/B table (51 MB) lives in the 192 MB L2 -> gathers are L2-resident.
// ---------------------------------------------------------------------------
__global__ void mlp_edge_score(const float* __restrict__ AB,
                               const int* __restrict__ src,
                               const int* __restrict__ dst,
                               const float* __restrict__ b1,
                               const float* __restrict__ W2,
                               const float* __restrict__ b2,
                               float* __restrict__ out,
                               int n_edges) {
    int e    = (blockIdx.x * blockDim.x + threadIdx.x) >> 5;
    int lane = threadIdx.x & 31;
    if (e >= n_edges) return;

    int s = src[e];
    int d = dst[e];

    const float4 a  = ((const float4*)(AB + (size_t)s * 256))[lane];
    const float4 b  = ((const float4*)(AB + (size_t)d * 256 + HIDDEN))[lane];
    const float4 bb = ((const float4*)b1)[lane];
    const float4 w  = ((const float4*)W2)[lane];

    float acc = fmaxf(a.x + b.x + bb.x, 0.0f) * w.x
              + fmaxf(a.y + b.y + bb.y, 0.0f) * w.y
              + fmaxf(a.z + b.z + bb.z, 0.0f) * w.z
              + fmaxf(a.w + b.w + bb.w, 0.0f) * w.w;

    // wave32 butterfly reduction
    for (int off = 16; off > 0; off >>= 1)
        acc += __shfl_xor(acc, off, 32);

    if (lane == 0) out[e] = acc + b2[0];
}

// ---------------------------------------------------------------------------
// Fallback (only if workspace is too small): direct per-edge compute.
// ---------------------------------------------------------------------------
__global__ void mlp_edge_direct(const float* __restrict__ h,
                                const int* __restrict__ src,
                                const int* __restrict__ dst,
                                const float* __restrict__ W1,
                                const float* __restrict__ b1,
                                const float* __restrict__ W2,
                                const float* __restrict__ b2,
                                float* __restrict__ out,
                                int n_edges) {
    int e = blockIdx.x;
    int k = threadIdx.x;  // 0..127
    if (e >= n_edges) return;
    const float* hs = h + (size_t)src[e] * D_FEAT;
    const float* hd = h + (size_t)dst[e] * D_FEAT;
    const float* w  = W1 + (size_t)k * (2 * D_FEAT);
    float acc = b1[k];
    for (int d = 0; d < D_FEAT; ++d)
        acc += hs[d] * w[d] + hd[d] * w[D_FEAT + d];
    acc = fmaxf(acc, 0.0f) * W2[k];
    __shared__ float red[HIDDEN];
    red[k] = acc;
    __syncthreads();
    for (int s2 = HIDDEN / 2; s2 > 0; s2 >>= 1) {
        if (k < s2) red[k] += red[k + s2];
        __syncthreads();
    }
    if (k == 0) out[e] = red[0] + b2[0];
}

// ---------------------------------------------------------------------------
extern "C" void kernel_launch(void* const* d_in, const int* in_sizes, int n_in,
                              void* d_out, int out_size, void* d_ws, size_t ws_size,
                              hipStream_t stream) {
    const float* h   = (const float*)d_in[0];
    const int*   src = (const int*)d_in[1];
    const int*   dst = (const int*)d_in[2];
    const float* W1  = (const float*)d_in[3];
    const float* b1  = (const float*)d_in[4];
    const float* W2  = (const float*)d_in[5];
    const float* b2  = (const float*)d_in[6];
    float* out = (float*)d_out;

    const int n_nodes = in_sizes[0] / D_FEAT;
    const int n_edges = in_sizes[1];

    const size_t ab_elems = (size_t)n_nodes * 256;
    const size_t wt_elems = (size_t)D_FEAT * 256;
    const size_t need     = (ab_elems + wt_elems) * sizeof(float);

    if (ws_size >= need) {
        float* AB = (float*)d_ws;
        float* Wt = AB + ab_elems;

        // Phase 0: weight transpose (tiny)
        mlp_prep_wt<<<(D_FEAT * 256 + 255) / 256, 256, 0, stream>>>(W1, Wt);

        // Phase 1: WMMA GEMM over all nodes
        int grid_m = (n_nodes + 15) / 16;
        mlp_node_gemm<<<grid_m, 256, 0, stream>>>(h, Wt, AB, n_nodes);

        // Phase 2: per-edge gather + relu + dot (one wave per edge)
        int edges_per_block = 256 / 32;
        int grid_e = (n_edges + edges_per_block - 1) / edges_per_block;
        mlp_edge_score<<<grid_e, 256, 0, stream>>>(AB, src, dst, b1, W2, b2, out, n_edges);
    } else {
        mlp_edge_direct<<<n_edges, HIDDEN, 0, stream>>>(h, src, dst, W1, b1, W2, b2,
                                                        out, n_edges);
    }
}